// CapsuleLayer_53446573032161
// MI455X (gfx1250) — compile-verified
//
#include <hip/hip_runtime.h>
#include <math.h>

// Problem constants (from the reference)
#define B_SZ 256
#define IC   1152
#define ID   8
#define OC   10
#define OD   16
#define ODIM (OC * OD)   // 160

typedef __attribute__((ext_vector_type(2))) float v2f;
typedef __attribute__((ext_vector_type(8))) float v8f;

// ---------------------------------------------------------------------------
// Kernel 1: u_hat[b,i,o,d] = sum_k W[i,o,d,k] * x[b,i,k]
// Per capsule i: GEMM [M=256 batches] x [K=8] x [N=160 (o,d)].
// One wave32 per (16-row batch tile, capsule). Two chained
// V_WMMA_F32_16X16X4_F32 cover K=8. 10 N-tiles per wave.
//
// A (16x4 f32) layout: lanes 0-15: M=lane, VGPR0=K0,VGPR1=K1;
//                      lanes 16-31: M=lane-16, VGPR0=K2,VGPR1=K3.
// B (4x16 f32) mirrors with N=lane&15.
// C/D (16x16 f32): VGPR r holds M=r (lanes 0-15) / M=r+8 (lanes 16-31), N=lane&15.
// ---------------------------------------------------------------------------
__global__ __launch_bounds__(32)
void uhat_wmma_kernel(const float* __restrict__ x,   // [B, IC, ID]
                      const float* __restrict__ W,   // [1, IC, OC, OD, ID]
                      float* __restrict__ uhat)      // [B, IC, OC*OD]
{
    const int mt   = blockIdx.x;        // batch tile 0..15
    const int i    = blockIdx.y;        // capsule 0..1151
    const int lane = threadIdx.x;       // 0..31 (wave32, EXEC all ones)
    const int kh   = lane >> 4;         // K half-select (0 or 1)
    const int l16  = lane & 15;

    // A tiles: x[mt*16 + l16, i, k]
    const int m = mt * 16 + l16;
    const float* xp = x + ((size_t)m * IC + i) * ID;
    v2f a0, a1;
    a0.x = xp[kh * 2 + 0]; a0.y = xp[kh * 2 + 1];   // K = 0..3
    a1.x = xp[kh * 2 + 4]; a1.y = xp[kh * 2 + 5];   // K = 4..7

    const float* Wi = W + (size_t)i * ODIM * ID;    // W[i, od, k]

    for (int nt = 0; nt < OC; ++nt) {
        const int n = nt * 16 + l16;                // od index
        const float* wp = Wi + (size_t)n * ID;
        v2f b0, b1;
        b0.x = wp[kh * 2 + 0]; b0.y = wp[kh * 2 + 1];
        b1.x = wp[kh * 2 + 4]; b1.y = wp[kh * 2 + 5];

        v8f c = {};
        // (neg_a, A, neg_b, B, c_mod, C, reuse_a, reuse_b)
        c = __builtin_amdgcn_wmma_f32_16x16x4_f32(false, a0, false, b0,
                                                  (short)0, c, false, false);
        c = __builtin_amdgcn_wmma_f32_16x16x4_f32(false, a1, false, b1,
                                                  (short)0, c, false, false);

        // Scatter D to u_hat[b, i, od]; keep in L2 (regular temporal stores):
        // u_hat (188.7 MB) fits the 192 MB L2 and is re-read by routing.
        #pragma unroll
        for (int r = 0; r < 8; ++r) {
            const int bb = mt * 16 + r + 8 * kh;
            uhat[((size_t)bb * IC + i) * ODIM + n] = c[r];
        }
    }
}

// ---------------------------------------------------------------------------
// Kernel 2: dynamic routing, one block per batch element.
// LDS: logits[IC*OC] (45KB) + softmax stats + s/v  (~57 KB < 320 KB WGP LDS).
// ---------------------------------------------------------------------------
__global__ __launch_bounds__(256)
void routing_kernel(const float* __restrict__ uhat,  // [B, IC, OC*OD]
                    float* __restrict__ out)         // [B, OC, OD]
{
    const int b   = blockIdx.x;
    const int tid = threadIdx.x;

    __shared__ float blog[IC * OC];   // routing logits
    __shared__ float cmax[IC];        // per-i softmax max
    __shared__ float cden[IC];        // per-i softmax denom
    __shared__ float s[ODIM];
    __shared__ float v[ODIM];

    const float* U = uhat + (size_t)b * IC * ODIM;

    for (int idx = tid; idx < IC * OC; idx += 256) blog[idx] = 0.0f;
    __syncthreads();

    for (int r = 0; r < 3; ++r) {
        // softmax normalization terms per input capsule i
        for (int i = tid; i < IC; i += 256) {
            float mx = blog[i * OC];
            #pragma unroll
            for (int o = 1; o < OC; ++o) mx = fmaxf(mx, blog[i * OC + o]);
            float den = 0.0f;
            #pragma unroll
            for (int o = 0; o < OC; ++o) den += __expf(blog[i * OC + o] - mx);
            cmax[i] = mx;
            cden[i] = den;
        }
        if (tid < ODIM) s[tid] = 0.0f;
        __syncthreads();

        // s[o,d] = sum_i c[i,o] * u_hat[i,o,d]  (register partials, LDS reduce)
        for (int o = 0; o < OC; ++o) {
            float acc[OD];
            #pragma unroll
            for (int d = 0; d < OD; ++d) acc[d] = 0.0f;
            for (int i = tid; i < IC; i += 256) {
                const float c = __expf(blog[i * OC + o] - cmax[i]) / cden[i];
                const float* up = U + ((size_t)i * OC + o) * OD;
                #pragma unroll
                for (int d = 0; d < OD; ++d) acc[d] += c * up[d];
            }
            #pragma unroll
            for (int d = 0; d < OD; ++d) atomicAdd(&s[o * OD + d], acc[d]);
        }
        __syncthreads();

        // squash: v = (|s|^2 / (1+|s|^2)) * s / (|s| + 1e-8)
        if (tid < OC) {
            float n2 = 0.0f;
            #pragma unroll
            for (int d = 0; d < OD; ++d) { float t = s[tid * OD + d]; n2 += t * t; }
            const float n     = sqrtf(n2);
            const float scale = (n2 / (1.0f + n2)) / (n + 1e-8f);
            #pragma unroll
            for (int d = 0; d < OD; ++d) v[tid * OD + d] = scale * s[tid * OD + d];
        }
        __syncthreads();

        // agreement update
        if (r < 2) {
            for (int i = tid; i < IC; i += 256) {
                const float* up = U + (size_t)i * ODIM;
                #pragma unroll
                for (int o = 0; o < OC; ++o) {
                    float a = 0.0f;
                    #pragma unroll
                    for (int d = 0; d < OD; ++d) a += up[o * OD + d] * v[o * OD + d];
                    blog[i * OC + o] += a;
                }
            }
            __syncthreads();
        }
    }

    if (tid < ODIM) out[(size_t)b * ODIM + tid] = v[tid];
}

// ---------------------------------------------------------------------------
extern "C" void kernel_launch(void* const* d_in, const int* in_sizes, int n_in,
                              void* d_out, int out_size, void* d_ws, size_t ws_size,
                              hipStream_t stream)
{
    (void)in_sizes; (void)n_in; (void)out_size; (void)ws_size;

    const float* x = (const float*)d_in[0];   // [256, 1152, 8]
    const float* W = (const float*)d_in[1];   // [1, 1152, 10, 16, 8]
    float* out  = (float*)d_out;              // [256, 10, 16]
    float* uhat = (float*)d_ws;               // needs 256*1152*160*4 = 188,743,680 B

    dim3 g1(B_SZ / 16, IC);                   // 16 batch tiles x 1152 capsules
    uhat_wmma_kernel<<<g1, 32, 0, stream>>>(x, W, uhat);

    routing_kernel<<<B_SZ, 256, 0, stream>>>(uhat, out);
}